// FeatureVoxel4D_37941741093419
// MI455X (gfx1250) — compile-verified
//
#include <hip/hip_runtime.h>
#include <math.h>

// FeatureVoxel4D (K-planes style) quadlinear interpolation, collapsed form:
//   out[n,c] = mask * sum_f ( bil(tx)[f,c]*bil(yz)[f,c]
//                           + bil(ty)[f,c]*bil(xz)[f,c]
//                           + bil(tz)[f,c]*bil(xy)[f,c] )
// Exact factorization of the reference's 16-corner weighted sum
// (quadlinear interp of a product of bilinear factors == product of
//  bilinear interps), giving 6x fewer FLOPs at identical gather traffic.
//
// Roofline: all six planes (~57 MB) stay resident in the 192 MB L2; the
// binding resource is ~6 KB of L2 gathers per point (~1.2 GB total), so the
// kernel is built around coalesced b128 gathers, not matrix ops (WMMA cannot
// express these contractions: the channel index appears on both operands).
//
// Lane mapping (wave32): half-wave (16 lanes) = 1 point; lane r = f*4 + cg,
// cg = 4-channel group -> every gather is a b128 load; each (plane,corner)
// is one vmem instruction per wave covering 2 points.
//
// CDNA5 specifics: positions staged via async DMA to LDS
// (global_load_async_to_lds_b128 + s_wait_asynccnt, wave-local, no barrier);
// output written with non-temporal stores so the 12.8 MB write-once stream
// does not displace the plane working set in L2.

#define FRANK 4
#define CCH   16

typedef __attribute__((ext_vector_type(4))) float vfloat4;

__device__ __forceinline__ float4 bilerp4(const float* __restrict__ plane,
                                          int D, int f, int cg,
                                          int ia, int ib, float fa, float fb) {
    const int sJ = CCH;            // stride over second plane index (in floats)
    const int sI = D * CCH;        // stride over first plane index
    const long long sF = (long long)D * D * CCH;
    const float* p00 = plane + (long long)f * sF + (long long)ia * sI + ib * sJ + cg * 4;
    const float4 v00 = *reinterpret_cast<const float4*>(p00);
    const float4 v01 = *reinterpret_cast<const float4*>(p00 + sJ);
    const float4 v10 = *reinterpret_cast<const float4*>(p00 + sI);
    const float4 v11 = *reinterpret_cast<const float4*>(p00 + sI + sJ);
    const float w00 = (1.0f - fa) * (1.0f - fb);
    const float w01 = (1.0f - fa) * fb;
    const float w10 = fa * (1.0f - fb);
    const float w11 = fa * fb;
    float4 o;
    o.x = w00 * v00.x + w01 * v01.x + w10 * v10.x + w11 * v11.x;
    o.y = w00 * v00.y + w01 * v01.y + w10 * v10.y + w11 * v11.y;
    o.z = w00 * v00.z + w01 * v01.z + w10 * v10.z + w11 * v11.z;
    o.w = w00 * v00.w + w01 * v01.w + w10 * v10.w + w11 * v11.w;
    return o;
}

__global__ __launch_bounds__(256)
void fv4d_interp_kernel(const float* __restrict__ positions,
                        const float* __restrict__ ptx, const float* __restrict__ pty,
                        const float* __restrict__ ptz, const float* __restrict__ pxy,
                        const float* __restrict__ pxz, const float* __restrict__ pyz,
                        const float* __restrict__ coord_range,
                        const float* __restrict__ resolution,
                        float* __restrict__ out, int N, int D) {
    __shared__ float pos_s[16 * 4];                 // one float4 per point slot

    const int t    = threadIdx.x;
    const int hw   = t >> 4;                        // point slot in block: 0..15
    const int r    = t & 15;                        // lane within point group
    const int fidx = r >> 2;                        // rank index 0..3
    const int cg   = r & 3;                         // channel group 0..3
    const long long n = (long long)blockIdx.x * 16 + hw;

    // ---- CDNA5 async DMA: stage this wave's 2 positions into LDS ----------
    if (r == 0 && n < (long long)N) {
        unsigned lds_off = (unsigned)(size_t)(&pos_s[hw * 4]);
        unsigned long long gaddr =
            (unsigned long long)(size_t)(positions + n * 4);
        asm volatile("global_load_async_to_lds_b128 %0, %1, off"
                     :: "v"(lds_off), "v"(gaddr) : "memory");
    }
    asm volatile("s_wait_asynccnt 0x0" ::: "memory");   // wave-local completion

    if (n >= (long long)N) return;

    // ---- normalize, mask, base/frac per dimension (t,x,y,z) ---------------
    int   bi[4];
    float fr[4];
    bool  ok = true;
    #pragma unroll
    for (int d = 0; d < 4; ++d) {
        const float lo  = coord_range[d];
        const float hi  = coord_range[4 + d];
        const float v   = (pos_s[hw * 4 + d] - lo) / (hi - lo);
        ok = ok && (v >= 0.0f) && (v < 1.0f);
        const float res = resolution[d];
        const int   ri  = (int)res;
        const float pv  = v * res;
        int b = (int)floorf(pv);
        b = min(max(b, 0), ri - 1);
        bi[d] = b;
        fr[d] = pv - (float)b;
    }

    // ---- six bilinear plane interpolations (collapsed corner sum) ---------
    const float4 itx = bilerp4(ptx, D, fidx, cg, bi[0], bi[1], fr[0], fr[1]);
    const float4 iyz = bilerp4(pyz, D, fidx, cg, bi[2], bi[3], fr[2], fr[3]);
    const float4 ity = bilerp4(pty, D, fidx, cg, bi[0], bi[2], fr[0], fr[2]);
    const float4 ixz = bilerp4(pxz, D, fidx, cg, bi[1], bi[3], fr[1], fr[3]);
    const float4 itz = bilerp4(ptz, D, fidx, cg, bi[0], bi[3], fr[0], fr[3]);
    const float4 ixy = bilerp4(pxy, D, fidx, cg, bi[1], bi[2], fr[1], fr[2]);

    float4 acc;
    acc.x = itx.x * iyz.x + ity.x * ixz.x + itz.x * ixy.x;
    acc.y = itx.y * iyz.y + ity.y * ixz.y + itz.y * ixy.y;
    acc.z = itx.z * iyz.z + ity.z * ixz.z + itz.z * ixy.z;
    acc.w = itx.w * iyz.w + ity.w * ixz.w + itz.w * ixy.w;

    // ---- reduce over rank f: lanes r, r^4, r^8 within the 16-lane group ---
    acc.x += __shfl_xor(acc.x, 4, 32);
    acc.y += __shfl_xor(acc.y, 4, 32);
    acc.z += __shfl_xor(acc.z, 4, 32);
    acc.w += __shfl_xor(acc.w, 4, 32);
    acc.x += __shfl_xor(acc.x, 8, 32);
    acc.y += __shfl_xor(acc.y, 8, 32);
    acc.z += __shfl_xor(acc.z, 8, 32);
    acc.w += __shfl_xor(acc.w, 8, 32);

    // ---- masked, coalesced, non-temporal store: 4 lanes x b128 per point --
    if (fidx == 0) {
        const float m = ok ? 1.0f : 0.0f;
        vfloat4 o;
        o.x = acc.x * m; o.y = acc.y * m; o.z = acc.z * m; o.w = acc.w * m;
        __builtin_nontemporal_store(
            o, reinterpret_cast<vfloat4*>(out + n * CCH + cg * 4));
    }
}

extern "C" void kernel_launch(void* const* d_in, const int* in_sizes, int n_in,
                              void* d_out, int out_size, void* d_ws, size_t ws_size,
                              hipStream_t stream) {
    const float* positions   = (const float*)d_in[0];
    const float* ptx         = (const float*)d_in[1];
    const float* pty         = (const float*)d_in[2];
    const float* ptz         = (const float*)d_in[3];
    const float* pxy         = (const float*)d_in[4];
    const float* pxz         = (const float*)d_in[5];
    const float* pyz         = (const float*)d_in[6];
    const float* coord_range = (const float*)d_in[7];
    const float* resolution  = (const float*)d_in[8];
    float* out = (float*)d_out;

    const int N = in_sizes[0] / 4;                       // positions: [N,4]
    // plane flat size = F * D * D * C  ->  D
    const long long dd = (long long)in_sizes[1] / (FRANK * CCH);
    const int D = (int)(sqrt((double)dd) + 0.5);

    const int tiles = (N + 15) / 16;                     // 16 points per block
    fv4d_interp_kernel<<<dim3(tiles), dim3(256), 0, stream>>>(
        positions, ptx, pty, ptz, pxy, pxz, pyz,
        coord_range, resolution, out, N, D);
}